// GraphEncoder_5231270166804
// MI455X (gfx1250) — compile-verified
//
#include <hip/hip_runtime.h>
#include <hip/hip_bf16.h>

#define NN 20000
#define EE 320000
#define EPN (EE + NN)      // edges + self loops
#define HID 256
#define HEADS 4
#define CH 64
#define LAYERS 3
#define NGRAPH 64
#define LN_EPS 1e-5f
#define NEG_SLOPE 0.2f

typedef __attribute__((ext_vector_type(2))) float v2f;
typedef __attribute__((ext_vector_type(8))) float v8f;

static __device__ __forceinline__ void atomicMaxF32(float* addr, float val) {
  // Works because: non-negative floats order like signed ints; negative floats
  // order inversely as unsigned ints; mixed cases resolve correctly from -inf init.
  if (val >= 0.0f) atomicMax((int*)addr, __float_as_int(val));
  else             atomicMin((unsigned int*)addr, __float_as_uint(val));
}

// ---------------- embedding: h = x @ W_emb + b_emb ----------------
__global__ __launch_bounds__(256) void k_embed(const float* __restrict__ x,
                                               const float* __restrict__ W,
                                               const float* __restrict__ b,
                                               float* __restrict__ h) {
  int gid = blockIdx.x * blockDim.x + threadIdx.x;
  if (gid >= NN * HID) return;
  int n = gid >> 8, c = gid & 255;
  float acc = b[c];
  acc += x[n * 3 + 0] * W[0 * HID + c];
  acc += x[n * 3 + 1] * W[1 * HID + c];
  acc += x[n * 3 + 2] * W[2 * HID + c];
  h[gid] = acc;
}

// ---------------- GEMM: C[N,256] = A[N,256] @ B[256,256] via WMMA f32 16x16x4 --
// One wave computes a 16(M) x 64(N) tile: 4 accumulators, A registers reused x4.
__global__ __launch_bounds__(256) void k_gemm(const float* __restrict__ A,
                                              const float* __restrict__ B,
                                              float* __restrict__ C) {
  int gid  = blockIdx.x * blockDim.x + threadIdx.x;
  int wv   = gid >> 5;          // global wave id: 1250 M-tiles x 4 col groups
  int lane = gid & 31;
  int tileM = wv >> 2;
  int cg    = wv & 3;
  int m16   = lane & 15;
  int kh    = lane >> 4;        // 0: K pair {0,1}, 1: K pair {2,3}
  int row   = tileM * 16 + m16;
  int n0    = cg * 64 + m16;    // B/C column for this lane (per 16-wide subtile)

  v8f acc0 = {}, acc1 = {}, acc2 = {}, acc3 = {};
  const float* __restrict__ arow = A + (size_t)row * HID;

  for (int k0 = 0; k0 < HID; k0 += 4) {
    int ka = k0 + kh * 2;
    v2f a;
    a.x = arow[ka];
    a.y = arow[ka + 1];
    const float* __restrict__ b0p = B + (size_t)ka * HID;
    const float* __restrict__ b1p = b0p + HID;
    v2f b;
    b.x = b0p[n0];      b.y = b1p[n0];
    acc0 = __builtin_amdgcn_wmma_f32_16x16x4_f32(false, a, false, b, (short)0, acc0, false, false);
    b.x = b0p[n0 + 16]; b.y = b1p[n0 + 16];
    acc1 = __builtin_amdgcn_wmma_f32_16x16x4_f32(false, a, false, b, (short)0, acc1, false, false);
    b.x = b0p[n0 + 32]; b.y = b1p[n0 + 32];
    acc2 = __builtin_amdgcn_wmma_f32_16x16x4_f32(false, a, false, b, (short)0, acc2, false, false);
    b.x = b0p[n0 + 48]; b.y = b1p[n0 + 48];
    acc3 = __builtin_amdgcn_wmma_f32_16x16x4_f32(false, a, false, b, (short)0, acc3, false, false);
  }

  // C/D layout: VGPR v -> row m_base + v (lanes 0-15) or m_base + v + 8 (lanes 16-31)
  int rbase = tileM * 16 + (kh << 3);
  int col   = cg * 64 + m16;
#pragma unroll
  for (int v = 0; v < 8; ++v) {
    float* __restrict__ crow = C + (size_t)(rbase + v) * HID + col;
    crow[0]  = acc0[v];
    crow[16] = acc1[v];
    crow[32] = acc2[v];
    crow[48] = acc3[v];
  }
}

// ---------------- per-(node,head) attention logits alpha_s / alpha_d ----------
__global__ __launch_bounds__(256) void k_alpha(const float* __restrict__ hp,
                                               const float* __restrict__ a_s,
                                               const float* __restrict__ a_d,
                                               float* __restrict__ alpha_s,
                                               float* __restrict__ alpha_d) {
  int gid = blockIdx.x * blockDim.x + threadIdx.x;
  if (gid >= NN * HEADS) return;
  int n = gid >> 2, hd = gid & 3;
  const float* __restrict__ v  = hp  + (size_t)n * HID + hd * CH;
  const float* __restrict__ ws = a_s + hd * CH;
  const float* __restrict__ wd = a_d + hd * CH;
  float accs = 0.f, accd = 0.f;
#pragma unroll 8
  for (int c = 0; c < CH; ++c) {
    float x = v[c];
    accs += x * ws[c];
    accd += x * wd[c];
  }
  alpha_s[gid] = accs;
  alpha_d[gid] = accd;
}

// ---------------- per-layer init: zero accumulator, m=-inf, denom=0 -----------
__global__ __launch_bounds__(256) void k_init_layer(float* __restrict__ acc,
                                                    float* __restrict__ m,
                                                    float* __restrict__ denom) {
  int gid = blockIdx.x * blockDim.x + threadIdx.x;
  if (gid < NN * HID) acc[gid] = 0.f;
  if (gid < NN * HEADS) {
    m[gid] = -__builtin_inff();
    denom[gid] = 0.f;
  }
}

// ---------------- edge pass 1: e = leaky_relu(as[src]+ad[dst]); segment max ---
__global__ __launch_bounds__(256) void k_edge_max(const int* __restrict__ ei,
                                                  const float* __restrict__ alpha_s,
                                                  const float* __restrict__ alpha_d,
                                                  float* __restrict__ m,
                                                  float* __restrict__ ebuf) {
  int gid = blockIdx.x * blockDim.x + threadIdx.x;
  if (gid >= EPN * HEADS) return;
  int eidx = gid >> 2, hd = gid & 3;
  int s, d; bool masked = false;
  if (eidx < EE) {
    s = ei[eidx]; d = ei[EE + eidx];
    masked = (s == d);            // pre-existing self-loops removed (PyG semantics)
  } else {
    s = d = eidx - EE;            // re-added self loops
  }
  float e;
  if (masked) {
    e = -__builtin_inff();
  } else {
    e = alpha_s[s * HEADS + hd] + alpha_d[d * HEADS + hd];
    e = (e > 0.f) ? e : NEG_SLOPE * e;
    atomicMaxF32(&m[d * HEADS + hd], e);
  }
  ebuf[gid] = e;
}

// ---------------- edge pass 2: ex = exp(e - m[dst]); segment sum --------------
__global__ __launch_bounds__(256) void k_edge_exp(const int* __restrict__ ei,
                                                  const float* __restrict__ m,
                                                  float* __restrict__ denom,
                                                  float* __restrict__ ebuf) {
  int gid = blockIdx.x * blockDim.x + threadIdx.x;
  if (gid >= EPN * HEADS) return;
  int eidx = gid >> 2, hd = gid & 3;
  int d = (eidx < EE) ? ei[EE + eidx] : (eidx - EE);
  float ex = expf(ebuf[gid] - m[d * HEADS + hd]);   // exp(-inf - finite) = 0
  atomicAdd(&denom[d * HEADS + hd], ex);
  ebuf[gid] = ex;
}

// ---------------- edge pass 3: alpha = ex / max(denom,1e-16) (in place) -------
__global__ __launch_bounds__(256) void k_edge_norm(const int* __restrict__ ei,
                                                   const float* __restrict__ denom,
                                                   float* __restrict__ ebuf) {
  int gid = blockIdx.x * blockDim.x + threadIdx.x;
  if (gid >= EPN * HEADS) return;
  int eidx = gid >> 2, hd = gid & 3;
  int d = (eidx < EE) ? ei[EE + eidx] : (eidx - EE);
  ebuf[gid] = ebuf[gid] / fmaxf(denom[d * HEADS + hd], 1e-16f);
}

// ---------------- scatter: acc[dst] += alpha * hp[src]; one wave per edge -----
__global__ __launch_bounds__(256) void k_scatter(const int* __restrict__ ei,
                                                 const float* __restrict__ alpha,
                                                 const float* __restrict__ hp,
                                                 float* __restrict__ acc) {
  int gid  = blockIdx.x * blockDim.x + threadIdx.x;
  int wv   = gid >> 5;
  if (wv >= EPN) return;          // wave-uniform exit
  int lane = gid & 31;
  int s, d;
  if (wv < EE) { s = ei[wv]; d = ei[EE + wv]; }
  else         { s = d = wv - EE; }
  int hd = lane >> 3;             // 8 contiguous channels per lane, 64 per head
  float al = alpha[wv * HEADS + hd];
  if (al == 0.f) return;          // masked self-loop duplicates: all-zero alpha
  const float4* __restrict__ sp = (const float4*)(hp + (size_t)s * HID + lane * 8);
  float4 v0 = sp[0], v1 = sp[1];
  float* __restrict__ dp = acc + (size_t)d * HID + lane * 8;
  atomicAdd(dp + 0, al * v0.x);
  atomicAdd(dp + 1, al * v0.y);
  atomicAdd(dp + 2, al * v0.z);
  atomicAdd(dp + 3, al * v0.w);
  atomicAdd(dp + 4, al * v1.x);
  atomicAdd(dp + 5, al * v1.y);
  atomicAdd(dp + 6, al * v1.z);
  atomicAdd(dp + 7, al * v1.w);
}

// ---------------- bias + LayerNorm; one wave per node -------------------------
__global__ __launch_bounds__(256) void k_layernorm(const float* __restrict__ acc,
                                                   const float* __restrict__ bg,
                                                   const float* __restrict__ g,
                                                   const float* __restrict__ bln,
                                                   float* __restrict__ out) {
  int gid  = blockIdx.x * blockDim.x + threadIdx.x;
  int wv   = gid >> 5;
  if (wv >= NN) return;
  int lane = gid & 31;
  const float4* __restrict__ ap = (const float4*)(acc + (size_t)wv * HID + lane * 8);
  float4 a0 = ap[0], a1 = ap[1];
  float v[8] = {a0.x, a0.y, a0.z, a0.w, a1.x, a1.y, a1.z, a1.w};
  float s = 0.f, ss = 0.f;
#pragma unroll
  for (int j = 0; j < 8; ++j) {
    v[j] += bg[lane * 8 + j];
    s  += v[j];
    ss += v[j] * v[j];
  }
#pragma unroll
  for (int off = 16; off > 0; off >>= 1) {
    s  += __shfl_xor(s,  off, 32);
    ss += __shfl_xor(ss, off, 32);
  }
  float mu  = s * (1.0f / HID);
  float var = ss * (1.0f / HID) - mu * mu;
  float rs  = rsqrtf(var + LN_EPS);
  float* __restrict__ op = out + (size_t)wv * HID + lane * 8;
#pragma unroll
  for (int j = 0; j < 8; ++j) {
    int ch = lane * 8 + j;
    op[j] = (v[j] - mu) * rs * g[ch] + bln[ch];
  }
}

// ---------------- global mean pool ----------------
__global__ __launch_bounds__(256) void k_pool_zero(float* __restrict__ sums,
                                                   float* __restrict__ cnts) {
  int gid = blockIdx.x * blockDim.x + threadIdx.x;
  if (gid < NGRAPH * HID) sums[gid] = 0.f;
  if (gid < NGRAPH) cnts[gid] = 0.f;
}

__global__ __launch_bounds__(256) void k_pool_acc(const float* __restrict__ h,
                                                  const int* __restrict__ batch,
                                                  float* __restrict__ sums,
                                                  float* __restrict__ cnts) {
  int gid = blockIdx.x * blockDim.x + threadIdx.x;
  if (gid >= NN * HID) return;
  int n = gid >> 8, c = gid & 255;
  int gph = batch[n];
  atomicAdd(&sums[gph * HID + c], h[gid]);
  if (c == 0) atomicAdd(&cnts[gph], 1.0f);
}

__global__ __launch_bounds__(256) void k_pool_fin(const float* __restrict__ sums,
                                                  const float* __restrict__ cnts,
                                                  float* __restrict__ out) {
  int gid = blockIdx.x * blockDim.x + threadIdx.x;
  if (gid >= NGRAPH * HID) return;
  out[gid] = sums[gid] / fmaxf(cnts[gid >> 8], 1.0f);
}

static inline int cdiv(long long a, long long b) { return (int)((a + b - 1) / b); }

extern "C" void kernel_launch(void* const* d_in, const int* in_sizes, int n_in,
                              void* d_out, int out_size, void* d_ws, size_t ws_size,
                              hipStream_t stream) {
  const float* x      = (const float*)d_in[0];
  const int*   ei     = (const int*)  d_in[1];   // [2, E]
  const int*   batch  = (const int*)  d_in[2];
  const float* W_emb  = (const float*)d_in[3];
  const float* b_emb  = (const float*)d_in[4];
  const float* W_gat  = (const float*)d_in[5];   // [L, 256, 256]
  const float* att_s  = (const float*)d_in[6];   // [L, 4, 64]
  const float* att_d  = (const float*)d_in[7];
  const float* b_gat  = (const float*)d_in[8];   // [L, 256]
  const float* ln_g   = (const float*)d_in[9];
  const float* ln_b   = (const float*)d_in[10];

  float* out_h    = (float*)d_out;                       // [N, 256]
  float* out_g    = out_h + (size_t)NN * HID;            // [64, 256]
  float* out_attn = out_g + (size_t)NGRAPH * HID;        // [3, E+N, 4]

  float* ws    = (float*)d_ws;
  float* hA    = ws;                                     // h / accumulator, N*256
  float* hp    = hA   + (size_t)NN * HID;                // projected feats, N*256
  float* as_   = hp   + (size_t)NN * HID;                // N*4
  float* ad_   = as_  + (size_t)NN * HEADS;              // N*4
  float* mbuf  = ad_  + (size_t)NN * HEADS;              // N*4
  float* dbuf  = mbuf + (size_t)NN * HEADS;              // N*4
  float* sums  = dbuf + (size_t)NN * HEADS;              // 64*256
  float* cnts  = sums + (size_t)NGRAPH * HID;            // 64

  const int T = 256;
  // 1. embedding
  k_embed<<<cdiv((long long)NN * HID, T), T, 0, stream>>>(x, W_emb, b_emb, hA);

  // 2. GAT layers
  for (int l = 0; l < LAYERS; ++l) {
    const float* Wl  = W_gat + (size_t)l * HID * HID;
    const float* asl = att_s + (size_t)l * HEADS * CH;
    const float* adl = att_d + (size_t)l * HEADS * CH;
    const float* bgl = b_gat + (size_t)l * HID;
    const float* gl  = ln_g  + (size_t)l * HID;
    const float* bll = ln_b  + (size_t)l * HID;
    float* attn_l = out_attn + (size_t)l * EPN * HEADS;

    // hp = hA @ Wl   (WMMA f32 16x16x4; 1250 M-tiles x 4 col groups = 5000 waves)
    k_gemm<<<(1250 * 4 * 32) / T, T, 0, stream>>>(hA, Wl, hp);
    // per-node attention logits
    k_alpha<<<cdiv((long long)NN * HEADS, T), T, 0, stream>>>(hp, asl, adl, as_, ad_);
    // reset accumulator (reuse hA) + softmax state
    k_init_layer<<<cdiv((long long)NN * HID, T), T, 0, stream>>>(hA, mbuf, dbuf);
    // segment softmax over destination neighborhoods (alpha staged in attn_l)
    k_edge_max <<<cdiv((long long)EPN * HEADS, T), T, 0, stream>>>(ei, as_, ad_, mbuf, attn_l);
    k_edge_exp <<<cdiv((long long)EPN * HEADS, T), T, 0, stream>>>(ei, mbuf, dbuf, attn_l);
    k_edge_norm<<<cdiv((long long)EPN * HEADS, T), T, 0, stream>>>(ei, dbuf, attn_l);
    // weighted scatter-add of source features (one wave per edge)
    k_scatter<<<cdiv((long long)EPN * 32, T), T, 0, stream>>>(ei, attn_l, hp, hA);
    // bias + LayerNorm -> next h (last layer writes the h output directly)
    float* hout = (l == LAYERS - 1) ? out_h : hA;
    k_layernorm<<<cdiv((long long)NN * 32, T), T, 0, stream>>>(hA, bgl, gl, bll, hout);
  }

  // 3. global mean pool
  k_pool_zero<<<cdiv((long long)NGRAPH * HID, T), T, 0, stream>>>(sums, cnts);
  k_pool_acc <<<cdiv((long long)NN * HID, T), T, 0, stream>>>(out_h, batch, sums, cnts);
  k_pool_fin <<<cdiv((long long)NGRAPH * HID, T), T, 0, stream>>>(sums, cnts, out_g);
}